// CRF2_33543694582322
// MI455X (gfx1250) — compile-verified
//
#include <hip/hip_runtime.h>
#include <hip/hip_bf16.h>
#include <math.h>

// Problem constants (from reference)
#define T_LEN     512
#define EMB_DIM   300
#define HID       512
#define NTAGS     32
#define START_TAG 30
#define STOP_TAG  31
#define NEG_VAL  -10000.0f

typedef float v2f __attribute__((ext_vector_type(2)));
typedef float v8f __attribute__((ext_vector_type(8)));

// ---------------------------------------------------------------------------
// 1) Embedding gather: X[t][e] = emb[ids[t]][e]
// ---------------------------------------------------------------------------
__global__ void embed_kernel(const int* __restrict__ ids,
                             const float* __restrict__ emb,
                             float* __restrict__ X) {
  int idx = blockIdx.x * blockDim.x + threadIdx.x;
  if (idx >= T_LEN * EMB_DIM) return;
  int t = idx / EMB_DIM;
  int e = idx - t * EMB_DIM;
  X[idx] = emb[(long)ids[t] * EMB_DIM + e];
}

// ---------------------------------------------------------------------------
// 2) WMMA GEMM (fp32):  Out[t][j] = sum_k X[t][k] * W[j][k]  (+ b0[j] + b1[j])
//    Register-blocked 4(M) x 2(N) 16x16 tiles per wave, software-pipelined:
//      - per K-step: 4 A-frags + 2 B-frags -> 8 independent WMMAs
//      - next step's fragments are loaded BEFORE this step's WMMAs, so the
//        s_wait before the WMMA group covers loads issued a full step earlier
//    Fragment layout per CDNA5 ISA 7.12.2 (V_WMMA_F32_16X16X4_F32):
//      A (16x4 f32, 2 VGPR): lanes 0-15 -> K=0,1 ; lanes 16-31 -> K=2,3
//      B (4x16 f32, 2 VGPR): mirrored
//      C/D (8 VGPR): VGPR r -> row r (lanes<16) / row r+8 (lanes>=16), col = lane&15
// ---------------------------------------------------------------------------
__device__ __forceinline__
void wmma8(const v2f a[4], const v2f b[2], v8f c[4][2]) {
#pragma unroll
  for (int m = 0; m < 4; ++m)
#pragma unroll
    for (int n = 0; n < 2; ++n)
      // 8 args: (neg_a, A, neg_b, B, c_mod, C, reuse_a, reuse_b)
      c[m][n] = __builtin_amdgcn_wmma_f32_16x16x4_f32(false, a[m], false, b[n],
                                                      (short)0, c[m][n], false, false);
}

__global__ __launch_bounds__(256)
void gemm_wmma(const float* __restrict__ X, int ldx, int K,
               const float* __restrict__ W,
               const float* __restrict__ b0, const float* __restrict__ b1,
               float* __restrict__ Out, int ldo, int nBlocksN, int nWaves) {
  int wave = (int)((blockIdx.x * blockDim.x + threadIdx.x) >> 5);
  if (wave >= nWaves) return;               // wave-uniform -> EXEC stays all-ones
  int lane = threadIdx.x & 31;
  int half = lane >> 4;                     // 0: K=0,1 / rows 0-7 ; 1: K=2,3 / rows 8-15
  int lm   = lane & 15;

  int mb = wave / nBlocksN, nb = wave % nBlocksN;
  int t0 = mb * 64, j0 = nb * 32;           // 4 M-tiles x 2 N-tiles

  v8f c[4][2];
#pragma unroll
  for (int n = 0; n < 2; ++n) {
    float bv = 0.0f;                        // bias depends only on column
    if (b0) bv += b0[j0 + 16 * n + lm];
    if (b1) bv += b1[j0 + 16 * n + lm];
#pragma unroll
    for (int m = 0; m < 4; ++m)
#pragma unroll
      for (int r = 0; r < 8; ++r) c[m][n][r] = bv;
  }

  const float* xp = X + (long)(t0 + lm) * ldx + 2 * half;  // 8B aligned (ldx in {300,1024})
  const float* wp = W + (long)(j0 + lm) * K   + 2 * half;
  const long xstep = (long)16 * ldx;
  const long wstep = (long)16 * K;

  // Prologue: fragments for k = 0
  v2f aC[4], bC[2];
#pragma unroll
  for (int n = 0; n < 2; ++n) bC[n] = *(const v2f*)(wp + n * wstep);
#pragma unroll
  for (int m = 0; m < 4; ++m) aC[m] = *(const v2f*)(xp + m * xstep);

  // Pipelined main loop: prefetch k+4 fragments, then 8 WMMAs on current.
  // (K is a multiple of 4 and >= 8 in all call sites; last step peeled below.)
#pragma unroll 2
  for (int k0 = 0; k0 < K - 4; k0 += 4) {
    v2f aN[4], bN[2];
#pragma unroll
    for (int n = 0; n < 2; ++n) bN[n] = *(const v2f*)(wp + n * wstep + k0 + 4);
#pragma unroll
    for (int m = 0; m < 4; ++m) aN[m] = *(const v2f*)(xp + m * xstep + k0 + 4);
    wmma8(aC, bC, c);
#pragma unroll
    for (int n = 0; n < 2; ++n) bC[n] = bN[n];   // rotate (renamed away by unroll)
#pragma unroll
    for (int m = 0; m < 4; ++m) aC[m] = aN[m];
  }
  wmma8(aC, bC, c);                         // epilogue step

#pragma unroll
  for (int m = 0; m < 4; ++m)
#pragma unroll
    for (int n = 0; n < 2; ++n)
#pragma unroll
      for (int r = 0; r < 8; ++r)
        Out[(long)(t0 + 16 * m + r + 8 * half) * ldo + j0 + 16 * n + lm] = c[m][n][r];
}

// ---------------------------------------------------------------------------
// 3) LSTM recurrence (one direction per workgroup; blockIdx.x = dir).
//    G already holds x_t @ Wih^T + bih + bhh for every t.
//    1024 threads: each computes 2 gate rows (dot of Whh row with h in LDS),
//    then threads 0..511 apply the cell update.  Whh streams from L2 each step.
//    (Kept on one WGP per direction: a grid-wide sync per step would cost more
//     than the ~1K FMAs/thread it could parallelize.)
// ---------------------------------------------------------------------------
__device__ __forceinline__ float sigf(float x) { return 1.0f / (1.0f + __expf(-x)); }

__global__ __launch_bounds__(1024)
void lstm_scan(const float* __restrict__ Gf, const float* __restrict__ Gb,
               const float* __restrict__ Wf, const float* __restrict__ Wb,
               float* __restrict__ Hout) {
  __shared__ float h_s[HID];
  __shared__ float g_s[4 * HID];

  const int dir = blockIdx.x;               // 0 = forward, 1 = backward
  const float* __restrict__ G   = dir ? Gb : Gf;
  const float* __restrict__ Whh = dir ? Wb : Wf;
  const int colOff = dir * HID;

  const int tid = threadIdx.x;
  float cstate = 0.0f;
  if (tid < HID) h_s[tid] = 0.0f;
  __syncthreads();

  for (int step = 0; step < T_LEN; ++step) {
    const int t = dir ? (T_LEN - 1 - step) : step;
    const float4* hv = (const float4*)h_s;
#pragma unroll
    for (int rr = 0; rr < 2; ++rr) {
      const int j = tid + rr * 1024;        // gate row 0..2047
      const float4* w = (const float4*)(Whh + (long)j * HID);
      float acc = G[(long)t * (4 * HID) + j];
#pragma unroll 4
      for (int k = 0; k < HID / 4; ++k) {
        float4 wv = w[k], hh = hv[k];
        acc += wv.x * hh.x + wv.y * hh.y + wv.z * hh.z + wv.w * hh.w;
      }
      g_s[j] = acc;
    }
    __syncthreads();
    if (tid < HID) {
      float ig = sigf (g_s[tid]);
      float fg = sigf (g_s[HID + tid]);
      float gg = tanhf(g_s[2 * HID + tid]);
      float og = sigf (g_s[3 * HID + tid]);
      cstate = fg * cstate + ig * gg;
      float hh = og * tanhf(cstate);
      h_s[tid] = hh;
      Hout[(long)t * (2 * HID) + colOff + tid] = hh;
    }
    __syncthreads();
  }
}

// ---------------------------------------------------------------------------
// 4) Viterbi: one wave32, one lane per tag.
// ---------------------------------------------------------------------------
__global__ __launch_bounds__(32)
void viterbi_kernel(const float* __restrict__ feats, const float* __restrict__ trans,
                    int* __restrict__ bp, float* __restrict__ out) {
  __shared__ float fvs[NTAGS];
  __shared__ float terms[NTAGS];
  const int lane = threadIdx.x;

  float tcol[NTAGS];                        // trans[i][lane], i = previous tag
#pragma unroll
  for (int i = 0; i < NTAGS; ++i) tcol[i] = trans[i * NTAGS + lane];

  fvs[lane] = (lane == START_TAG) ? 0.0f : NEG_VAL;
  __syncthreads();

  for (int t = 0; t < T_LEN; ++t) {
    float best = -3.4e38f; int arg = 0;
#pragma unroll
    for (int i = 0; i < NTAGS; ++i) {
      float s = fvs[i] + tcol[i];
      if (s > best) { best = s; arg = i; }  // strict '>' keeps first max (argmax semantics)
    }
    bp[t * NTAGS + lane] = arg;
    float nv = best + feats[t * NTAGS + lane];
    __syncthreads();                        // all reads of fvs done
    fvs[lane] = nv;
    __syncthreads();
  }

  terms[lane] = fvs[lane] + trans[lane * NTAGS + STOP_TAG];
  __syncthreads();
  if (lane == 0) {
    float best = terms[0]; int bi = 0;
    for (int i = 1; i < NTAGS; ++i) if (terms[i] > best) { best = terms[i]; bi = i; }
    out[0] = best;                          // score
    int cur = bi;
    out[T_LEN] = (float)cur;                // path[T-1] -> out[1 + (T-1)]
    for (int t = T_LEN - 1; t > 0; --t) {
      cur = bp[t * NTAGS + cur];
      out[t] = (float)cur;                  // path[t-1] -> out[1 + (t-1)]
    }
  }
}

// ---------------------------------------------------------------------------
// Host side
// ---------------------------------------------------------------------------
static void launch_gemm(const float* X, int ldx, int K, const float* W,
                        const float* b0, const float* b1,
                        float* O, int ldo, int tilesM, int tilesN, hipStream_t s) {
  int blocksM = tilesM / 4;                 // 4 M-tiles per wave
  int blocksN = tilesN / 2;                 // 2 N-tiles per wave
  int waves = blocksM * blocksN;
  int threads = waves * 32;
  gemm_wmma<<<(threads + 255) / 256, 256, 0, s>>>(X, ldx, K, W, b0, b1, O, ldo, blocksN, waves);
}

extern "C" void kernel_launch(void* const* d_in, const int* in_sizes, int n_in,
                              void* d_out, int out_size, void* d_ws, size_t ws_size,
                              hipStream_t stream) {
  (void)in_sizes; (void)n_in; (void)out_size; (void)ws_size;

  const int*   feats = (const int*)d_in[0];
  const float* emb   = (const float*)d_in[1];
  const float* P[16];                       // l{0,1}d{0,1}: Wih, Whh, bih, bhh
  for (int i = 0; i < 16; ++i) P[i] = (const float*)d_in[2 + i];
  const float* W_out = (const float*)d_in[18];
  const float* b_out = (const float*)d_in[19];
  const float* trans = (const float*)d_in[20];
  float* out = (float*)d_out;

  // Workspace carve-up (~12.7 MB, all 16B-aligned offsets)
  char* ws = (char*)d_ws;
  float* X0 = (float*)ws; ws += sizeof(float) * T_LEN * EMB_DIM;      // 600 KB
  float* Gf = (float*)ws; ws += sizeof(float) * T_LEN * 4 * HID;      // 4 MB
  float* Gb = (float*)ws; ws += sizeof(float) * T_LEN * 4 * HID;      // 4 MB
  float* H0 = (float*)ws; ws += sizeof(float) * T_LEN * 2 * HID;      // 2 MB
  float* H1 = (float*)ws; ws += sizeof(float) * T_LEN * 2 * HID;      // 2 MB
  float* FT = (float*)ws; ws += sizeof(float) * T_LEN * NTAGS;        // 64 KB
  int*   BP = (int*)ws;   ws += sizeof(int)   * T_LEN * NTAGS;        // 64 KB

  // 1) Embedding gather
  {
    int n = T_LEN * EMB_DIM;
    embed_kernel<<<(n + 255) / 256, 256, 0, stream>>>(feats, emb, X0);
  }

  const int tilesM = T_LEN / 16;            // 32
  const int tilesG = (4 * HID) / 16;        // 128

  // 2) Layer 0: input projections (WMMA) then recurrence (both dirs in one launch)
  launch_gemm(X0, EMB_DIM, EMB_DIM, P[0], P[2],  P[3],  Gf, 4 * HID, tilesM, tilesG, stream);
  launch_gemm(X0, EMB_DIM, EMB_DIM, P[4], P[6],  P[7],  Gb, 4 * HID, tilesM, tilesG, stream);
  lstm_scan<<<2, 1024, 0, stream>>>(Gf, Gb, P[1], P[5], H0);

  // 3) Layer 1
  launch_gemm(H0, 2 * HID, 2 * HID, P[8],  P[10], P[11], Gf, 4 * HID, tilesM, tilesG, stream);
  launch_gemm(H0, 2 * HID, 2 * HID, P[12], P[14], P[15], Gb, 4 * HID, tilesM, tilesG, stream);
  lstm_scan<<<2, 1024, 0, stream>>>(Gf, Gb, P[9], P[13], H1);

  // 4) Tag projection (WMMA, 32x2 tiles -> 8 waves = one 256-thread block)
  launch_gemm(H1, 2 * HID, 2 * HID, W_out, b_out, nullptr, FT, NTAGS, tilesM, NTAGS / 16, stream);

  // 5) Viterbi decode (single wave32)
  viterbi_kernel<<<1, 32, 0, stream>>>(FT, trans, BP, out);
}